// VocabAttention_82669530514111
// MI455X (gfx1250) — compile-verified
//
#include <hip/hip_runtime.h>

// VocabAttention for MI455X (gfx1250, wave32).
// Roofline: E (50257x300 fp32 = 60MB) read exactly twice (att pass + alpha@E
// pass) -> ~120MB @ 23.3TB/s ~ 5.2us floor. att phase is elementwise-bound
// (3 VALU ops per of 241M elements, VOPD dual-issued) -> VALU with q/W in LDS.
// The two true matmuls (query GEMM, alpha@E GEMM) use V_WMMA_F32_16X16X4_F32.

#define BATCH 16
#define VDIM  300
#define DDIM  512
#define VOCAB 50257
#define QS    304          // padded row stride for query/acc in ws
#define NT    19           // ceil(VDIM/16) n-tiles
#define KSLABS 512         // K-partitions for alpha@E (waves in flight for 60MB stream)

// ws layout (float index)
#define WS_Q    0
#define WS_ACC  (BATCH*QS)          // 4864
#define WS_MAX  (2*BATCH*QS)        // 9728 (stored as ordered uint32)
#define WS_SUM  (2*BATCH*QS + BATCH)// 9744
// total 9760 floats = 39040 bytes of d_ws

typedef float v2f __attribute__((ext_vector_type(2)));
typedef float v8f __attribute__((ext_vector_type(8)));

__device__ __forceinline__ unsigned f32_ord(float f) {
    unsigned u = __float_as_uint(f);
    return (u & 0x80000000u) ? ~u : (u | 0x80000000u);
}
__device__ __forceinline__ float ord_f32(unsigned u) {
    return __uint_as_float((u & 0x80000000u) ? (u & 0x7fffffffu) : ~u);
}

// ---------------------------------------------------------------- init scratch
__global__ void k_init(float* ws) {
    int t = threadIdx.x;
    for (int i = t; i < BATCH * QS; i += 256) ws[WS_ACC + i] = 0.0f;
    if (t < BATCH) {
        ((unsigned*)ws)[WS_MAX + t] = 0u;   // ordered-uint minimum
        ws[WS_SUM + t] = 0.0f;
    }
}

// ------------------------------------------- query = dh @ W_dec^T + b_dec (WMMA)
// One wave per 16-wide d-tile. C(16b x 16d) = A(16b x 512k) * B(512k x 16d),
// B[k][d] = W_dec[d][k]. K stepped by 4 with v_wmma_f32_16x16x4_f32.
__global__ __launch_bounds__(32) void k_query(const float* __restrict__ dh,
                                              const float* __restrict__ Wd,
                                              const float* __restrict__ bd,
                                              float* __restrict__ ws) {
    const int lane = threadIdx.x;
    const int half = lane >> 4;          // 0: K+0/K+1, 1: K+2/K+3 (ISA A/B layout)
    const int l    = lane & 15;
    const int n0   = blockIdx.x * 16;
    int drow = n0 + l; if (drow > VDIM - 1) drow = VDIM - 1;   // clamp (no divergence)

    const float* dhrow = dh + l * DDIM;       // A row (M = batch = l)
    const float* wrow  = Wd + drow * DDIM;    // B col n=l  <- W_dec row drow

    v8f c = {0.f,0.f,0.f,0.f,0.f,0.f,0.f,0.f};
    for (int k0 = 0; k0 < DDIM; k0 += 4) {
        v2f a, b;
        a[0] = dhrow[k0 + 2*half];
        a[1] = dhrow[k0 + 2*half + 1];
        b[0] = wrow [k0 + 2*half];
        b[1] = wrow [k0 + 2*half + 1];
        c = __builtin_amdgcn_wmma_f32_16x16x4_f32(false, a, false, b,
                                                  (short)0, c, false, false);
    }
    const int d = n0 + l;
    if (d < VDIM) {
        const float bias = bd[d];
#pragma unroll
        for (int r = 0; r < 8; ++r) {
            const int brow = r + 8*half;       // C layout: vgpr r -> M=r / M=r+8
            ws[WS_Q + brow * QS + d] = c[r] + bias;
        }
    }
}

// ------------------- att[b,v] = sum_d relu(E[v,d]+q[b,d])*W[d]; track max per b
__global__ __launch_bounds__(256) void k_att(const float* __restrict__ E,
                                             const float* __restrict__ Wf,
                                             const float* __restrict__ ws,
                                             float* __restrict__ alpha,
                                             unsigned* __restrict__ gmax) {
    __shared__ __align__(16) float qs[BATCH * QS];
    __shared__ __align__(16) float wls[QS];
    __shared__ unsigned smax[BATCH];
    const int tid = threadIdx.x;
    for (int i = tid; i < BATCH * QS; i += 256) qs[i] = ws[WS_Q + i];
    for (int i = tid; i < VDIM;       i += 256) wls[i] = Wf[i];
    if (tid < BATCH) smax[tid] = 0u;
    __syncthreads();

    const int v = blockIdx.x * 256 + tid;
    const bool valid = v < VOCAB;
    float acc[BATCH];
#pragma unroll
    for (int b = 0; b < BATCH; ++b) acc[b] = 0.0f;

    if (valid) {
        const float4* erow = (const float4*)(E + (long)v * VDIM);
        for (int dq = 0; dq < VDIM / 4; ++dq) {       // 75 float4 chunks, exact
            __builtin_prefetch(erow + dq + 16, 0, 0); // stream ~256B ahead (global_prefetch_b8)
            const float4 e4 = erow[dq];
            const float4 w4 = ((const float4*)wls)[dq];
#pragma unroll
            for (int b = 0; b < BATCH; ++b) {
                const float4 q4 = ((const float4*)(qs + b * QS))[dq];
                acc[b] += fmaxf(e4.x + q4.x, 0.0f) * w4.x;
                acc[b] += fmaxf(e4.y + q4.y, 0.0f) * w4.y;
                acc[b] += fmaxf(e4.z + q4.z, 0.0f) * w4.z;
                acc[b] += fmaxf(e4.w + q4.w, 0.0f) * w4.w;
            }
        }
    }
#pragma unroll
    for (int b = 0; b < BATCH; ++b) {
        unsigned u = 0u;
        if (valid) {
            alpha[b * VOCAB + v] = acc[b];            // raw att staged in d_out
            u = f32_ord(acc[b]);
        }
        for (int off = 16; off > 0; off >>= 1) {
            unsigned o = __shfl_down(u, off);
            u = (o > u) ? o : u;
        }
        if ((tid & 31) == 0) atomicMax(&smax[b], u);
    }
    __syncthreads();
    if (tid < BATCH) atomicMax(&gmax[tid], smax[tid]);
}

// ---------------------------- e = exp(att - max); accumulate per-batch sums
__global__ __launch_bounds__(256) void k_exp(float* __restrict__ alpha,
                                             const unsigned* __restrict__ gmax,
                                             float* __restrict__ gsum) {
    __shared__ float smaxf[BATCH];
    __shared__ float ssum[BATCH];
    const int tid = threadIdx.x;
    if (tid < BATCH) { smaxf[tid] = ord_f32(gmax[tid]); ssum[tid] = 0.0f; }
    __syncthreads();

    const int v = blockIdx.x * 256 + tid;
    const bool valid = v < VOCAB;
#pragma unroll
    for (int b = 0; b < BATCH; ++b) {
        float e = 0.0f;
        if (valid) {
            e = __expf(alpha[b * VOCAB + v] - smaxf[b]);
            alpha[b * VOCAB + v] = e;                 // unnormalized alpha
        }
        for (int off = 16; off > 0; off >>= 1) e += __shfl_down(e, off);
        if ((tid & 31) == 0) atomicAdd(&ssum[b], e);
    }
    __syncthreads();
    if (tid < BATCH) atomicAdd(&gsum[tid], ssum[tid]);
}

// --------------- acc[b,d] += sum_{v in slab} e[b,v]*E[v,d]  (WMMA, K=vocab)
// One wave per K-slab; 19 n-tile accumulators held live so A (e) is loaded
// once per K-step and E is streamed exactly once across all blocks.
__global__ __launch_bounds__(32) void k_wsum(const float* __restrict__ E,
                                             const float* __restrict__ alpha,
                                             float* __restrict__ ws) {
    const int lane = threadIdx.x;
    const int half = lane >> 4;
    const int l    = lane & 15;
    const int nslabs = gridDim.x;
    const int slab   = ((VOCAB + nslabs - 1) / nslabs + 3) & ~3;
    const int kbeg   = blockIdx.x * slab;
    const int kend   = min(kbeg + slab, VOCAB);

    const v8f zero = {0.f,0.f,0.f,0.f,0.f,0.f,0.f,0.f};
    v8f c[NT];
#pragma unroll
    for (int t = 0; t < NT; ++t) c[t] = zero;

    for (int k0 = kbeg; k0 < kend; k0 += 4) {
        const int ka  = k0 + 2*half;
        const int ka1 = ka + 1;
        v2f a;
        a[0] = (ka  < VOCAB) ? alpha[(long)l * VOCAB + ka ] : 0.0f;  // A[m=l][k]
        a[1] = (ka1 < VOCAB) ? alpha[(long)l * VOCAB + ka1] : 0.0f;
        const int kb  = min(ka,  VOCAB - 1);
        const int kb1 = min(ka1, VOCAB - 1);
        const float* eb0 = E + (long)kb  * VDIM;
        const float* eb1 = E + (long)kb1 * VDIM;
#pragma unroll
        for (int t = 0; t < NT; ++t) {
            int col = t * 16 + l; if (col > VDIM - 1) col = VDIM - 1;
            v2f b;
            b[0] = eb0[col];                           // B[k][n] = E[v=k][d=n]
            b[1] = eb1[col];
            c[t] = __builtin_amdgcn_wmma_f32_16x16x4_f32(false, a, false, b,
                                                         (short)0, c[t], false, false);
        }
    }
#pragma unroll
    for (int t = 0; t < NT; ++t) {
        const int d = t * 16 + l;
        if (d < VDIM) {
#pragma unroll
            for (int r = 0; r < 8; ++r) {
                const int brow = r + 8*half;
                atomicAdd(&ws[WS_ACC + brow * QS + d], c[t][r]);
            }
        }
    }
}

// ---------------------------- normalize: alpha /= sum ; awe = acc / sum
__global__ __launch_bounds__(256) void k_final(const float* __restrict__ ws,
                                               float* __restrict__ awe,
                                               float* __restrict__ alpha) {
    const int b = blockIdx.y;
    const int v = blockIdx.x * 256 + threadIdx.x;
    const float inv = 1.0f / ws[WS_SUM + b];
    if (v < VOCAB) alpha[b * VOCAB + v] *= inv;
    if (v < VDIM)  awe[b * VDIM + v] = ws[WS_ACC + b * QS + v] * inv;
}

extern "C" void kernel_launch(void* const* d_in, const int* in_sizes, int n_in,
                              void* d_out, int out_size, void* d_ws, size_t ws_size,
                              hipStream_t stream) {
    const float* dh = (const float*)d_in[0];   // (16,512)
    const float* E  = (const float*)d_in[1];   // (50257,300)
    const float* Wd = (const float*)d_in[2];   // (300,512)
    const float* bd = (const float*)d_in[3];   // (300,)
    const float* Wf = (const float*)d_in[4];   // (300,)
    // d_in[5] = b_full: constant shift per row -> softmax-invariant, skipped.

    float* out   = (float*)d_out;
    float* awe   = out;                        // 16*300
    float* alpha = out + BATCH * VDIM;         // 16*50257
    float* ws    = (float*)d_ws;               // needs 39040 bytes

    const int vb = (VOCAB + 255) / 256;        // 197

    k_init <<<1, 256, 0, stream>>>(ws);
    k_query<<<NT, 32, 0, stream>>>(dh, Wd, bd, ws);
    k_att  <<<vb, 256, 0, stream>>>(E, Wf, ws, alpha, (unsigned*)ws + WS_MAX);
    k_exp  <<<vb, 256, 0, stream>>>(alpha, (const unsigned*)ws + WS_MAX, ws + WS_SUM);
    k_wsum <<<KSLABS, 32, 0, stream>>>(E, alpha, ws);
    k_final<<<dim3(vb, BATCH), 256, 0, stream>>>(ws, awe, alpha);
}